// PtModule_76166950027701
// MI455X (gfx1250) — compile-verified
//
#include <hip/hip_runtime.h>
#include <hip/hip_bf16.h>

typedef float v4f __attribute__((ext_vector_type(4)));
typedef float v2f __attribute__((ext_vector_type(2)));

// Packed fp32 FMA: one VOP3P instruction computes two lanes of d = a*b + c.
// 64-bit "v" operands allocate aligned VGPR pairs as VOP3P requires.
__device__ inline v2f pk_fma(v2f a, v2f b, v2f c) {
    v2f d;
    asm("v_pk_fma_f32 %0, %1, %2, %3" : "=v"(d) : "v"(a), "v"(b), "v"(c));
    return d;
}

__device__ inline v4f affine4(v4f x, v2f k_mul, v2f k_add) {
    v2f lo = {x.x, x.y};
    v2f hi = {x.z, x.w};
    lo = pk_fma(lo, k_mul, k_add);
    hi = pk_fma(hi, k_mul, k_add);
    v4f y = {lo.x, lo.y, hi.x, hi.y};
    return y;
}

// y = 1.5*x + 3  (== ((x+2)*3)/2), streamed with 128-bit NT loads/stores.
// 2-way unrolled grid-stride loop: two b128 loads in flight per wave before
// the first dependent compute, halving loop-control overhead per byte.
__global__ __launch_bounds__(256) void fused_affine_kernel(
    const v4f* __restrict__ in4, v4f* __restrict__ out4, unsigned n4,
    const float* __restrict__ in_s, float* __restrict__ out_s, unsigned n)
{
    const v2f k_mul = {1.5f, 1.5f};
    const v2f k_add = {3.0f, 3.0f};

    const unsigned stride = gridDim.x * blockDim.x;
    unsigned i = blockIdx.x * blockDim.x + threadIdx.x;

    // Main unrolled loop: process i and i+stride per trip (both coalesced).
    for (; i + stride < n4; i += 2u * stride) {
        v4f x0 = __builtin_nontemporal_load(&in4[i]);
        v4f x1 = __builtin_nontemporal_load(&in4[i + stride]);

        v4f y0 = affine4(x0, k_mul, k_add);
        __builtin_nontemporal_store(y0, &out4[i]);
        v4f y1 = affine4(x1, k_mul, k_add);
        __builtin_nontemporal_store(y1, &out4[i + stride]);
    }
    // Odd leftover float4 chunk for this thread.
    if (i < n4) {
        v4f x = __builtin_nontemporal_load(&in4[i]);
        v4f y = affine4(x, k_mul, k_add);
        __builtin_nontemporal_store(y, &out4[i]);
    }

    // Scalar tail for n not divisible by 4 (defensive; n = 32Mi is divisible).
    const unsigned base = n4 * 4u;
    for (unsigned j = base + blockIdx.x * blockDim.x + threadIdx.x; j < n; j += stride) {
        out_s[j] = fmaf(in_s[j], 1.5f, 3.0f);
    }
}

extern "C" void kernel_launch(void* const* d_in, const int* in_sizes, int n_in,
                              void* d_out, int out_size, void* d_ws, size_t ws_size,
                              hipStream_t stream)
{
    (void)n_in; (void)d_ws; (void)ws_size; (void)out_size;

    const float* in = (const float*)d_in[0];
    float* out = (float*)d_out;

    const unsigned n  = (unsigned)in_sizes[0];   // 4096*8192 = 33,554,432
    const unsigned n4 = n / 4u;                  // 8,388,608 float4s

    const int threads = 256;                     // 8 wave32 per block
    // Each thread handles ~4 float4s (2 unrolled x 2 grid-stride trips):
    // enough in-flight b128 loads per WGP to cover HBM latency at 23.3 TB/s,
    // small enough grid for fast launch.
    unsigned blocks = (n4 + threads * 4u - 1u) / (threads * 4u);
    if (blocks == 0) blocks = 1;

    fused_affine_kernel<<<blocks, threads, 0, stream>>>(
        (const v4f*)in, (v4f*)out, n4, in, out, n);
}